// MolNet_2783138808276
// MI455X (gfx1250) — compile-verified
//
#include <hip/hip_runtime.h>
#include <hip/hip_bf16.h>

// ---------------------------------------------------------------------------
// GCN forward for MI455X (gfx1250, wave32).
// Dense transforms use V_WMMA_F32_16X16X4_F32 (exact fp32). Edge aggregation
// uses float atomics into L2-resident buffers (12.8MB << 192MB L2).
// ---------------------------------------------------------------------------

typedef float v2f __attribute__((ext_vector_type(2)));
typedef float v8f __attribute__((ext_vector_type(8)));

#define NNODES  200000
#define NEDGES  6400000
#define NFEAT   128
#define NHID    16
#define NMLP    100
#define NCLS    27
#define NGRAPH  4096

// ---------------- simple utility kernels ----------------

__global__ void fill_f32(float* __restrict__ p, int n, float v) {
    int i = blockIdx.x * blockDim.x + threadIdx.x;
    if (i < n) p[i] = v;
}

// deg[d] += 1 for each edge (deg pre-filled with 1.0 for the self loop)
__global__ void deg_accum(const int* __restrict__ dst, float* __restrict__ deg, int e) {
    int i = blockIdx.x * blockDim.x + threadIdx.x;
    if (i < e) atomicAdd(&deg[dst[i]], 1.0f);
}

__global__ void rsqrt_ip(float* __restrict__ p, int n) {
    int i = blockIdx.x * blockDim.x + threadIdx.x;
    if (i < n) p[i] = rsqrtf(p[i]);
}

// ---------------- WMMA fp32 GEMM: T[M,16] = X[M,K] @ W[K,16] ----------------
// One wave per 16-row output tile. K stepped by 4 using V_WMMA_F32_16X16X4_F32.
// A layout (16x4 f32): lane m=lane&15; VGPR j holds K = 2*(lane>=16) + j.
// B layout (4x16 f32): lane n=lane&15; VGPR j holds K = 2*(lane>=16) + j.
// D layout (16x16 f32): VGPR v -> row v + 8*(lane>=16), col lane&15.
__global__ __launch_bounds__(256)
void gemm_wmma_f32(const float* __restrict__ X, const float* __restrict__ W,
                   float* __restrict__ T, int M, int K) {
    __shared__ float lw[NFEAT * NHID];           // up to 128x16 = 8KB
    const int tid = threadIdx.x;
    const int kn  = K * NHID;
    for (int i = tid; i < kn; i += 256) lw[i] = W[i];
    __syncthreads();

    const int wave = tid >> 5;
    const int lane = tid & 31;
    const int tile = blockIdx.x * 8 + wave;
    if (tile * 16 >= M) return;                  // wave-uniform: EXEC stays all-1s

    const int row   = tile * 16 + (lane & 15);
    const int kHalf = (lane >> 4) << 1;          // 0 or 2
    const int n     = lane & 15;

    v8f c = {};
    const float* xr = X + (size_t)row * K + kHalf;
    for (int k0 = 0; k0 < K; k0 += 4) {
        float2 av = *(const float2*)(xr + k0);   // b64 load, 8B aligned
        v2f a; a.x = av.x; a.y = av.y;
        v2f b;
        b.x = lw[(k0 + kHalf)     * NHID + n];
        b.y = lw[(k0 + kHalf + 1) * NHID + n];
        c = __builtin_amdgcn_wmma_f32_16x16x4_f32(
                /*neg_a=*/false, a, /*neg_b=*/false, b,
                /*c_mod=*/(short)0, c, /*reuse_a=*/false, /*reuse_b=*/false);
    }

    const int rbase = tile * 16 + ((lane >> 4) << 3);
#pragma unroll
    for (int v = 0; v < 8; ++v)
        T[(size_t)(rbase + v) * NHID + n] = c[v];
}

// ---------------- edge scatter: OUT[d] += T[s] * dinv[s]*dinv[d] ------------
// 4 threads per edge, each handles a float4 slice of the 16 features.
__global__ void edge_scatter(const int* __restrict__ src, const int* __restrict__ dst,
                             const float* __restrict__ dinv, const float* __restrict__ T,
                             float* __restrict__ OUT, int e) {
    int t  = blockIdx.x * blockDim.x + threadIdx.x;
    int ed = t >> 2, q = t & 3;
    if (ed >= e) return;
    int s = src[ed], d = dst[ed];
    float w = dinv[s] * dinv[d];
    const float4 m = *(const float4*)(T + (size_t)s * NHID + q * 4);
    float* o = OUT + (size_t)d * NHID + q * 4;
    atomicAdd(o + 0, m.x * w);
    atomicAdd(o + 1, m.y * w);
    atomicAdd(o + 2, m.z * w);
    atomicAdd(o + 3, m.w * w);
}

// self-loop contribution + bias (+ optional ReLU), in place on ACC
__global__ void finalize_node(float* __restrict__ ACC, const float* __restrict__ T,
                              const float* __restrict__ dinv, const float* __restrict__ bias,
                              int n16, int doRelu) {
    int idx = blockIdx.x * blockDim.x + threadIdx.x;
    if (idx >= n16) return;
    int i = idx >> 4, f = idx & 15;
    float di = dinv[i];
    float v  = ACC[idx] + T[idx] * di * di + bias[f];
    ACC[idx] = doRelu ? fmaxf(v, 0.0f) : v;
}

// global_add_pool: G[batch[i]] += H[i]
__global__ void pool_kernel(const int* __restrict__ bvec, const float* __restrict__ Hn,
                            float* __restrict__ Gp, int n) {
    int t = blockIdx.x * blockDim.x + threadIdx.x;
    int i = t >> 2, q = t & 3;
    if (i >= n) return;
    int g = bvec[i];
    const float4 h = *(const float4*)(Hn + (size_t)i * NHID + q * 4);
    float* o = Gp + (size_t)g * NHID + q * 4;
    atomicAdd(o + 0, h.x);
    atomicAdd(o + 1, h.y);
    atomicAdd(o + 2, h.z);
    atomicAdd(o + 3, h.w);
}

// g2 = relu(relu(g) @ Wl1 + bl1)   [4096,16] x [16,100]
__global__ void lin1_kernel(const float* __restrict__ G, const float* __restrict__ W,
                            const float* __restrict__ B, float* __restrict__ O) {
    int t = blockIdx.x * blockDim.x + threadIdx.x;
    if (t >= NGRAPH * NMLP) return;
    int gi = t / NMLP, j = t % NMLP;
    float acc = B[j];
#pragma unroll
    for (int k = 0; k < NHID; ++k)
        acc = fmaf(fmaxf(G[gi * NHID + k], 0.0f), W[k * NMLP + j], acc);
    O[t] = fmaxf(acc, 0.0f);
}

// out = g2 @ Wl2 + bl2   [4096,100] x [100,27]
__global__ void lin2_kernel(const float* __restrict__ G, const float* __restrict__ W,
                            const float* __restrict__ B, float* __restrict__ O) {
    int t = blockIdx.x * blockDim.x + threadIdx.x;
    if (t >= NGRAPH * NCLS) return;
    int gi = t / NCLS, c = t % NCLS;
    float acc = B[c];
    for (int k = 0; k < NMLP; ++k)
        acc = fmaf(G[gi * NMLP + k], W[k * NCLS + c], acc);
    O[t] = acc;
}

// ---------------------------------------------------------------------------

extern "C" void kernel_launch(void* const* d_in, const int* in_sizes, int n_in,
                              void* d_out, int out_size, void* d_ws, size_t ws_size,
                              hipStream_t stream) {
    const float* x   = (const float*)d_in[0];
    const int*  eidx = (const int*)d_in[1];
    const int*  bvec = (const int*)d_in[2];
    const float* W1  = (const float*)d_in[3];
    const float* b1  = (const float*)d_in[4];
    const float* W2  = (const float*)d_in[5];
    const float* b2  = (const float*)d_in[6];
    const float* Wl1 = (const float*)d_in[7];
    const float* bl1 = (const float*)d_in[8];
    const float* Wl2 = (const float*)d_in[9];
    const float* bl2 = (const float*)d_in[10];
    float* out = (float*)d_out;

    // workspace layout
    char* ws = (char*)d_ws;
    size_t off = 0;
    auto walloc = [&](size_t bytes) -> void* {
        void* p = ws + off;
        off = (off + bytes + 255) & ~(size_t)255;
        return p;
    };
    float* deg  = (float*)walloc((size_t)NNODES * 4);            // then dinv
    float* bufA = (float*)walloc((size_t)NNODES * NHID * 4);     // t1, later t2
    float* bufB = (float*)walloc((size_t)NNODES * NHID * 4);     // h1, later h2
    float* gbuf = (float*)walloc((size_t)NGRAPH * NHID * 4);
    float* g2   = (float*)walloc((size_t)NGRAPH * NMLP * 4);

    const int* src = eidx;
    const int* dst = eidx + NEDGES;

    const int TB = 256;
    const int nN    = (NNODES + TB - 1) / TB;
    const int nE    = (NEDGES + TB - 1) / TB;
    const int nN16  = (NNODES * NHID + TB - 1) / TB;
    const int nE4   = (NEDGES * 4 + TB - 1) / TB;
    const int nN4   = (NNODES * 4 + TB - 1) / TB;
    const int nTile = ((NNODES / 16) + 7) / 8;                   // 12500 tiles / 8 waves

    // degrees & normalization
    fill_f32<<<nN, TB, 0, stream>>>(deg, NNODES, 1.0f);          // self loop
    deg_accum<<<nE, TB, 0, stream>>>(dst, deg, NEDGES);
    rsqrt_ip<<<nN, TB, 0, stream>>>(deg, NNODES);                // deg -> dinv
    float* dinv = deg;

    // layer 1: t1 = x@W1 ; h1 = relu(scatter(t1) + selfloop + b1)
    gemm_wmma_f32<<<nTile, TB, 0, stream>>>(x, W1, bufA, NNODES, NFEAT);
    fill_f32<<<nN16, TB, 0, stream>>>(bufB, NNODES * NHID, 0.0f);
    edge_scatter<<<nE4, TB, 0, stream>>>(src, dst, dinv, bufA, bufB, NEDGES);
    finalize_node<<<nN16, TB, 0, stream>>>(bufB, bufA, dinv, b1, NNODES * NHID, 1);

    // layer 2: t2 = h1@W2 ; h2 = scatter(t2) + selfloop + b2  (no relu)
    gemm_wmma_f32<<<nTile, TB, 0, stream>>>(bufB, W2, bufA, NNODES, NHID);
    fill_f32<<<nN16, TB, 0, stream>>>(bufB, NNODES * NHID, 0.0f);
    edge_scatter<<<nE4, TB, 0, stream>>>(src, dst, dinv, bufA, bufB, NEDGES);
    finalize_node<<<nN16, TB, 0, stream>>>(bufB, bufA, dinv, b2, NNODES * NHID, 0);

    // pooling + MLP head
    fill_f32<<<(NGRAPH * NHID + TB - 1) / TB, TB, 0, stream>>>(gbuf, NGRAPH * NHID, 0.0f);
    pool_kernel<<<nN4, TB, 0, stream>>>(bvec, bufB, gbuf, NNODES);
    lin1_kernel<<<(NGRAPH * NMLP + TB - 1) / TB, TB, 0, stream>>>(gbuf, Wl1, bl1, g2);
    lin2_kernel<<<(NGRAPH * NCLS + TB - 1) / TB, TB, 0, stream>>>(g2, Wl2, bl2, out);
}